// FisherLayer_17763984736302
// MI455X (gfx1250) — compile-verified
//
#include <hip/hip_runtime.h>
#include <math.h>

// Problem constants (from reference)
#define NB      32
#define T_TOTAL 8192
#define DDIM    64
#define KMIX    64

#define TCH     128            // T rows per workgroup chunk
#define DPAD    68             // padded LDS row stride (floats); 68*4=272B = 17*16B -> float4-aligned rows
#define VLEN    129            // 1 + 2*D per (n,k)
#define OUTROW  (KMIX * VLEN)  // 8256 floats per n

typedef __attribute__((ext_vector_type(2))) float v2f;
typedef __attribute__((ext_vector_type(8))) float v8f;

// ---- gfx1250 async global->LDS path (ASYNCcnt), hedged on builtin presence ----
#if defined(__has_builtin)
#if __has_builtin(__builtin_amdgcn_global_load_async_to_lds_b128)
#define HAVE_ASYNC_LDS 1
#endif
#endif

// The async builtin takes pointers to 16B int vectors: AS1 source, AS3 dest.
typedef int v4i_ __attribute__((vector_size(4 * sizeof(int))));
#define GPTR128(p)                                    \
  ((__attribute__((address_space(1))) v4i_*)          \
   (__attribute__((address_space(1))) void*)(p))
#define LPTR128(p)                                    \
  ((__attribute__((address_space(3))) v4i_*)          \
   (__attribute__((address_space(3))) void*)(p))

__device__ __forceinline__ void wait_asynccnt0() {
#if defined(__has_builtin) && __has_builtin(__builtin_amdgcn_s_wait_asynccnt)
  __builtin_amdgcn_s_wait_asynccnt(0);
#else
  asm volatile("s_wait_asynccnt 0" ::: "memory");
#endif
}

// CDNA5 f32 WMMA: D[16x16] += A[16x4] * B[4x16], full fp32 path.
__device__ __forceinline__ v8f wmma_f32(v2f a, v2f b, v8f c) {
  return __builtin_amdgcn_wmma_f32_16x16x4_f32(
      /*neg_a=*/false, a, /*neg_b=*/false, b,
      /*c_mod=*/(short)0, c, /*reuse_a=*/false, /*reuse_b=*/false);
}

__global__ void zero_ws(float* ws, int n) {
  int i = blockIdx.x * blockDim.x + threadIdx.x;
  if (i < n) ws[i] = 0.0f;
}

// One workgroup handles one (n, 128-row T chunk): posterior GEMM -> softmax ->
// pooling GEMMs, all staged in LDS; stats accumulated via global f32 atomics.
__global__ __launch_bounds__(256) void gmm_stats(
    const float* __restrict__ x, const float* __restrict__ w,
    const float* __restrict__ means, const float* __restrict__ covars,
    float* __restrict__ s0g, float* __restrict__ s1g, float* __restrict__ s2g) {
  __shared__ float xs[TCH][DPAD];        // x chunk
  __shared__ float xq[TCH][DPAD];        // x^2 chunk
  __shared__ float pl[TCH][DPAD];        // logits, then posteriors p
  __shared__ float Wl[2 * DDIM][DPAD];   // [m*ic ; -0.5*ic] weight matrix
  __shared__ float Ab[KMIX];             // per-k bias

  const int tid  = threadIdx.x;
  const int lane = tid & 31;
  const int wv   = tid >> 5;   // 8 waves
  const int half = lane >> 4;  // lane half: 0 or 1
  const int mr   = lane & 15;
  const int n    = blockIdx.y;
  const int t0   = blockIdx.x * TCH;

  const float* xg = x + ((size_t)n * T_TOTAL + t0) * DDIM;

#ifdef HAVE_ASYNC_LDS
  // ---- kick off async DMA-style global->LDS staging of the x chunk ----
  // 2048 x b128 transfers, exactly 8 per thread: no divergence, EXEC stays full.
  for (int i = tid; i < TCH * (DDIM / 4); i += 256) {
    int row = i >> 4, d = (i & 15) * 4;
    __builtin_amdgcn_global_load_async_to_lds_b128(
        GPTR128(xg + row * DDIM + d), LPTR128(&xs[row][d]), 0, 0);
  }
#endif

  // ---- per-block GMM parameter prep (overlaps the async copy) ----
  if (tid < KMIX) {
    const int k = tid;
    float acc = 0.0f;
    for (int d = 0; d < DDIM; ++d) {
      float c  = covars[k * DDIM + d];
      float ic = 1.0f / c;
      float m  = means[k * DDIM + d];
      Wl[d][k]        = m * ic;
      Wl[DDIM + d][k] = -0.5f * ic;
      acc += logf(c) + m * m * ic;
    }
    Ab[k] = 2.0f * logf(w[k]) - 0.5f * acc;  // D*log2pi cancels in softmax
  }

#ifdef HAVE_ASYNC_LDS
  wait_asynccnt0();   // this wave's async LDS writes are done
  __syncthreads();    // all waves' writes visible
  // square from LDS (rows are 16B-aligned: 68 floats = 272B stride)
  for (int i = tid; i < TCH * (DDIM / 4); i += 256) {
    int row = i >> 4, d = (i & 15) * 4;
    float4 v = *(const float4*)&xs[row][d];
    xq[row][d + 0] = v.x * v.x; xq[row][d + 1] = v.y * v.y;
    xq[row][d + 2] = v.z * v.z; xq[row][d + 3] = v.w * v.w;
  }
#else
  // ---- fallback: stage x chunk through VGPRs (coalesced float4) ----
  {
    const float4* xg4 = (const float4*)xg;
    for (int i = tid; i < TCH * (DDIM / 4); i += 256) {
      int row = i >> 4, c4 = i & 15;
      float4 v = xg4[row * (DDIM / 4) + c4];
      int d = c4 * 4;
      xs[row][d + 0] = v.x;       xs[row][d + 1] = v.y;
      xs[row][d + 2] = v.z;       xs[row][d + 3] = v.w;
      xq[row][d + 0] = v.x * v.x; xq[row][d + 1] = v.y * v.y;
      xq[row][d + 2] = v.z * v.z; xq[row][d + 3] = v.w * v.w;
    }
  }
#endif
  __syncthreads();

  // ---- phase A: logits[128x64] = [x, x^2][128x128] * Wl[128x64] + bias ----
  // 8 Mtiles x 4 Ntiles = 32 tiles, 4 per wave; 32 k-steps of K=4 each.
  for (int tile = wv; tile < 32; tile += 8) {
    const int m0 = (tile >> 2) * 16;  // t offset
    const int n0 = (tile & 3) * 16;   // k offset
    v8f acc = {};
    for (int kk = 0; kk < 2 * DDIM; kk += 4) {
      const float(*As)[DPAD] =
          (kk < DDIM) ? (const float(*)[DPAD])xs : (const float(*)[DPAD])xq;
      const int da = (kk & (DDIM - 1)) + 2 * half;
      v2f a, b;
      // A 16x4 f32 layout: VGPR0 = K{0|2}, VGPR1 = K{1|3} split by lane half
      a.x = As[m0 + mr][da];
      a.y = As[m0 + mr][da + 1];
      // B 4x16: rows striped across lanes within a VGPR (mirror of A)
      b.x = Wl[kk + 2 * half][n0 + mr];
      b.y = Wl[kk + 2 * half + 1][n0 + mr];
      acc = wmma_f32(a, b, acc);
    }
    const float bias = Ab[n0 + mr];
    // C layout: VGPR r holds rows (r | r+8) split by lane half, col = mr
    for (int r = 0; r < 8; ++r)
      pl[m0 + r + 8 * half][n0 + mr] = acc[r] + bias;
  }
  __syncthreads();

  // ---- phase B: softmax over K per row ----
  if (tid < TCH) {
    float mx = -3.0e38f;
    for (int k = 0; k < KMIX; ++k) mx = fmaxf(mx, pl[tid][k]);
    float s = 0.0f;
    for (int k = 0; k < KMIX; ++k) {
      float e = expf(pl[tid][k] - mx);
      pl[tid][k] = e;
      s += e;
    }
    float inv = 1.0f / s;
    for (int k = 0; k < KMIX; ++k) pl[tid][k] *= inv;
  }
  __syncthreads();

  // ---- s0[k] += column sums of p ----
  if (tid < KMIX) {
    float s = 0.0f;
    for (int t = 0; t < TCH; ++t) s += pl[t][tid];
    atomicAdd(&s0g[n * KMIX + tid], s);
  }

  // ---- phase C: s1 += p^T * x, s2 += p^T * x^2 (two 64x64 GEMMs, inner=128) ----
  // 2 gemms x 16 tiles = 32 jobs, 4 per wave.
  for (int job = wv; job < 32; job += 8) {
    const int g   = job >> 4;          // 0 -> s1 (B=xs), 1 -> s2 (B=xq)
    const int tl  = job & 15;
    const int km0 = (tl >> 2) * 16;    // k offset (M)
    const int d0  = (tl & 3) * 16;     // d offset (N)
    const float(*Bs)[DPAD] =
        g ? (const float(*)[DPAD])xq : (const float(*)[DPAD])xs;
    v8f acc = {};
    for (int kk = 0; kk < TCH; kk += 4) {
      v2f a, b;
      // A[M=k, inner=t] = p[t][k]
      a.x = pl[kk + 2 * half][km0 + mr];
      a.y = pl[kk + 2 * half + 1][km0 + mr];
      b.x = Bs[kk + 2 * half][d0 + mr];
      b.y = Bs[kk + 2 * half + 1][d0 + mr];
      acc = wmma_f32(a, b, acc);
    }
    float* dst = (g ? s2g : s1g) + (size_t)n * KMIX * DDIM;
    for (int r = 0; r < 8; ++r) {
      int k = km0 + r + 8 * half;
      atomicAdd(&dst[k * DDIM + d0 + mr], acc[r]);
    }
  }
}

// One block per n: build v, signed-sqrt, L2-normalize.
__global__ __launch_bounds__(256) void fisher_finalize(
    const float* __restrict__ w, const float* __restrict__ means,
    const float* __restrict__ covars, const float* __restrict__ s0g,
    const float* __restrict__ s1g, const float* __restrict__ s2g,
    float* __restrict__ out) {
  __shared__ float vb[OUTROW];  // 33 KB
  __shared__ float ssum;
  const int n = blockIdx.x, tid = threadIdx.x;
  if (tid == 0) ssum = 0.0f;
  __syncthreads();

  float local = 0.0f;
  for (int i = tid; i < OUTROW; i += 256) {
    int k = i / VLEN, r = i % VLEN;
    float wk = w[k], sw = sqrtf(wk);
    float s0 = s0g[n * KMIX + k];
    float val;
    if (r == 0) {
      val = (s0 - (float)T_TOTAL * wk) / sw;
    } else if (r <= DDIM) {
      int d = r - 1;
      float m = means[k * DDIM + d], c = covars[k * DDIM + d];
      float s1 = s1g[((size_t)n * KMIX + k) * DDIM + d];
      val = (s1 - m * s0) / (sw * sqrtf(c));
    } else {
      int d = r - 1 - DDIM;
      float m = means[k * DDIM + d], c = covars[k * DDIM + d];
      float s1 = s1g[((size_t)n * KMIX + k) * DDIM + d];
      float s2 = s2g[((size_t)n * KMIX + k) * DDIM + d];
      val = (s2 - 2.0f * m * s1 + (m * m - c) * s0) / (1.41421356237f * sw * c);
    }
    float pv = copysignf(sqrtf(fabsf(val)), val);  // power-normalization
    vb[i] = pv;
    local += pv * pv;
  }
  atomicAdd(&ssum, local);  // ds_add_f32 reduction
  __syncthreads();
  float inv = 1.0f / sqrtf(ssum);
  for (int i = tid; i < OUTROW; i += 256)
    out[(size_t)n * OUTROW + i] = vb[i] * inv;
}

extern "C" void kernel_launch(void* const* d_in, const int* in_sizes, int n_in,
                              void* d_out, int out_size, void* d_ws, size_t ws_size,
                              hipStream_t stream) {
  const float* x      = (const float*)d_in[0];  // [N,T,D]
  const float* w      = (const float*)d_in[1];  // [K]
  const float* means  = (const float*)d_in[2];  // [K,D]
  const float* covars = (const float*)d_in[3];  // [K,D]
  float* out = (float*)d_out;

  // Workspace layout: s0 [N,K] | s1 [N,K,D] | s2 [N,K,D]  (~1.0 MB)
  float* s0 = (float*)d_ws;
  float* s1 = s0 + NB * KMIX;
  float* s2 = s1 + NB * KMIX * DDIM;
  const int nz = NB * KMIX + 2 * NB * KMIX * DDIM;

  zero_ws<<<(nz + 255) / 256, 256, 0, stream>>>(s0, nz);
  dim3 grid(T_TOTAL / TCH, NB);
  gmm_stats<<<grid, 256, 0, stream>>>(x, w, means, covars, s0, s1, s2);
  fisher_finalize<<<NB, 256, 0, stream>>>(w, means, covars, s0, s1, s2, out);
}